// SparseGraphLearn_781684048180
// MI455X (gfx1250) — compile-verified
//
#include <hip/hip_runtime.h>
#include <hip/hip_bf16.h>

typedef float v2f __attribute__((ext_vector_type(2)));
typedef float v8f __attribute__((ext_vector_type(8)));

#define IN_F  256
#define OUT_F 128

// ---------------------------------------------------------------------------
// GEMM: h[N,128] = X[N,256] @ W[256,128], fp32 WMMA 16x16x4.
// Block = 256 threads = 8 waves; each wave owns a 16-row strip and all 8
// N-tiles (8 v8f accumulators). Weight staged through LDS in two 64KB K-halves
// via the CDNA5 async global->LDS path (ASYNCcnt). A-frags prefetched in
// chunks of 8 so global-load latency is amortized over 64 WMMAs.
// ---------------------------------------------------------------------------
__global__ __launch_bounds__(256) void sgl_gemm_wmma(
    const float* __restrict__ X, const float* __restrict__ W,
    float* __restrict__ H, int N) {
  __shared__ float Wl[128 * OUT_F];  // 64 KB: one K-half of the weight

  const int lane  = threadIdx.x & 31;
  const int wave  = threadIdx.x >> 5;
  const int m     = lane & 15;        // row within tile / col within tile
  const int khalf = (lane >> 4) * 2;  // lanes 0-15 -> K{0,1}, 16-31 -> K{2,3}

  const int rowTile = blockIdx.x * 8 + wave;
  const int row0    = rowTile * 16;
  const bool active = row0 < N;                      // tiles are exact (N%16==0)
  const int row0c   = active ? row0 : (N >= 16 ? N - 16 : 0);  // clamp for loads

  const uint32_t lds0 = (uint32_t)(uintptr_t)(&Wl[0]);

  v8f acc[8] = {};  // 8 N-tiles of 16 -> full 128-wide strip

  for (int half = 0; half < 2; ++half) {
    __syncthreads();  // all waves done reading previous half
    {
      // Async DMA: global -> LDS, 16B per lane per issue, no VGPR round-trip.
      const float4* src = reinterpret_cast<const float4*>(W + half * 128 * OUT_F);
      for (int i = threadIdx.x; i < (128 * OUT_F) / 4; i += blockDim.x) {
        uint64_t ga = (uint64_t)(uintptr_t)(src + i);
        uint32_t la = lds0 + (uint32_t)i * 16u;
        asm volatile("global_load_async_to_lds_b128 %0, %1, off"
                     :: "v"(la), "v"(ga) : "memory");
      }
      asm volatile("s_wait_asynccnt 0" ::: "memory");
    }
    __syncthreads();

    // A-frag pointer: element (row0c+m, half*128 + khalf + kk)
    const v2f* xrow = reinterpret_cast<const v2f*>(
        X + (size_t)(row0c + m) * IN_F + half * 128 + khalf);

    for (int kc = 0; kc < 128; kc += 32) {
      // Prefetch 8 k-steps of A into registers: 8 global_load_b64 in flight.
      v2f af[8];
#pragma unroll
      for (int j = 0; j < 8; ++j) af[j] = xrow[(kc + j * 4) >> 1];

#pragma unroll
      for (int j = 0; j < 8; ++j) {
        const int kk = kc + j * 4;
#pragma unroll
        for (int t = 0; t < 8; ++t) {
          v2f bfrag;
          bfrag.x = Wl[(kk + khalf) * OUT_F + t * 16 + m];
          bfrag.y = Wl[(kk + khalf + 1) * OUT_F + t * 16 + m];
          acc[t] = __builtin_amdgcn_wmma_f32_16x16x4_f32(
              /*neg_a=*/false, af[j], /*neg_b=*/false, bfrag,
              /*c_mod=*/(short)0, acc[t], /*reuse_a=*/false, /*reuse_b=*/false);
        }
      }
    }
  }

  if (!active) return;
  // C/D layout: VGPR r -> M = r (+8 for lanes 16-31), N = lane&15
  const int mofs = (lane >> 4) * 8;
#pragma unroll
  for (int t = 0; t < 8; ++t)
#pragma unroll
    for (int r = 0; r < 8; ++r)
      H[(size_t)(row0 + mofs + r) * OUT_F + t * 16 + m] = acc[t][r];
}

// ---------------------------------------------------------------------------
// Edge stage: one wave32 per edge. Lane L handles features [4L, 4L+4):
// two coalesced float4 row reads (512B/row/wave, L2-resident since h=51MB
// fits the 192MB L2), fused |x-y|*a partial sums, butterfly reduce, ReLU.
// ---------------------------------------------------------------------------
__global__ __launch_bounds__(256) void sgl_edge_attn(
    const float* __restrict__ H, const long long* __restrict__ e0,
    const long long* __restrict__ e1, const float* __restrict__ a,
    float* __restrict__ out, long long E) {
  const int lane = threadIdx.x & 31;
  const int wavesPerBlock = blockDim.x >> 5;
  const long long wid    = (long long)blockIdx.x * wavesPerBlock + (threadIdx.x >> 5);
  const long long stride = (long long)gridDim.x * wavesPerBlock;

  const float4 av = reinterpret_cast<const float4*>(a)[lane];  // a in registers

  for (long long e = wid; e < E; e += stride) {
    const long long i0 = e0[e];
    const long long i1 = e1[e];
    const float4 x = reinterpret_cast<const float4*>(H + i0 * OUT_F)[lane];
    const float4 y = reinterpret_cast<const float4*>(H + i1 * OUT_F)[lane];
    float s = fabsf(x.x - y.x) * av.x;
    s = fmaf(fabsf(x.y - y.y), av.y, s);
    s = fmaf(fabsf(x.z - y.z), av.z, s);
    s = fmaf(fabsf(x.w - y.w), av.w, s);
#pragma unroll
    for (int off = 16; off > 0; off >>= 1)
      s += __shfl_xor(s, off, 32);  // wave32 butterfly
    if (lane == 0) out[e] = fmaxf(s, 0.0f);
  }
}

// ---------------------------------------------------------------------------
extern "C" void kernel_launch(void* const* d_in, const int* in_sizes, int n_in,
                              void* d_out, int out_size, void* d_ws, size_t ws_size,
                              hipStream_t stream) {
  const float*     inputs = (const float*)d_in[0];      // [N, 256]
  const long long* edge   = (const long long*)d_in[1];  // [2, E] int64
  const float*     weight = (const float*)d_in[2];      // [256, 128]
  const float*     a      = (const float*)d_in[3];      // [128, 1]

  const int N = in_sizes[0] / IN_F;          // 100000
  const long long E = in_sizes[1] / 2;       // 1600000

  float* h  = (float*)d_out;                 // output 0: [N,128]
  float* ew = (float*)d_out + (size_t)N * OUT_F;  // output 1: [E]

  // GEMM: 8 row-tiles (of 16) per block
  const int mTiles = (N + 15) / 16;
  const int gBlocks = (mTiles + 7) / 8;
  sgl_gemm_wmma<<<gBlocks, 256, 0, stream>>>(inputs, weight, h, N);

  // Edge attention: one wave per edge, 8 waves per block
  const int eBlocks = (int)((E + 7) / 8);
  sgl_edge_attn<<<eBlocks, 256, 0, stream>>>(h, edge, edge + E, a, ew, E);
}